// L1HyMixDe_36326833389612
// MI455X (gfx1250) — compile-verified
//
#include <hip/hip_runtime.h>

// L1HyMixDe on gfx1250 (MI455X).
//  - All matmuls via V_WMMA_F32_16X16X4_F32 (f32 problem, traffic-bound:
//    ~66 MB working set is L2-resident (192 MB), GEMM FLOPs ~15 GF total,
//    so full-precision f32 WMMA is the right matrix path).
//  - NN GEMMs: 4-wave workgroups sharing a 16-wide B panel staged into LDS
//    by the Tensor Data Mover (TENSOR_LOAD_TO_LDS + s_wait_tensorcnt),
//    double buffered. Fallback to synchronous LDS staging if the TDM
//    builtin is unavailable.
//  - NT (Gram) GEMMs: K-streaming with global_prefetch.

typedef __attribute__((ext_vector_type(2))) float v2f;
typedef __attribute__((ext_vector_type(8))) float v8f;
typedef __attribute__((ext_vector_type(4))) unsigned int u32x4;
typedef __attribute__((ext_vector_type(8))) int i32x8;
typedef __attribute__((ext_vector_type(4))) int i32x4;

#define NN 16384   // 128*128 pixels
#define NB 191     // bands
#define BP 192     // padded bands (12*16)
#define KP 16      // padded subspace dim (k=10)
#define KB 16      // K chunk staged in LDS per TDM descriptor
#define PI_F 3.14159265358979f

#if defined(__has_builtin)
# if __has_builtin(__builtin_amdgcn_tensor_load_to_lds) && __has_builtin(__builtin_amdgcn_s_wait_tensorcnt)
#  define USE_TDM 1
# endif
#endif
#ifndef USE_TDM
# define USE_TDM 0
#endif

// ---------------------------------------------------------------------------
// TDM: load a KB x 16 f32 tile (row stride ldb elements) from global into LDS.
// D# per CDNA5 ISA 8.3/8.4: group0 = {count=1, lds_addr, global_addr, type=2},
// group1 = {data_size=4B, tensor_dim0=16, tensor_dim1=KB, tile_dim0=16,
//           tile_dim1=KB, tensor_dim0_stride=ldb}. Groups 2/3 unused (2-D),
// 5th arg is the second 256-bit SGPR group of the 6-arg builtin (zeros).
// ---------------------------------------------------------------------------
__device__ __forceinline__ void tdm_load_tile(const float* gsrc, unsigned lds_byte, int ldb)
{
#if USE_TDM
    unsigned long long ga = (unsigned long long)(size_t)gsrc;
    u32x4 g0;
    g0[0] = 1u;                                                  // count=1
    g0[1] = lds_byte;                                            // lds_addr
    g0[2] = (unsigned)(ga & 0xffffffffu);                        // global_addr lo
    g0[3] = (unsigned)((ga >> 32) & 0x01ffffffu) | 0x80000000u;  // addr[56:32] | type=2
    i32x8 g1;
    g1[0] = 0x20000;       // data_size = 2 (4 bytes)
    g1[1] = 16 << 16;      // tensor_dim0 = 16 (bits 79:48 low half)
    g1[2] = KB << 16;      // tensor_dim1 = KB (bits 111:80 low half)
    g1[3] = 16 << 16;      // tile_dim0 = 16  (bits 127:112)
    g1[4] = KB;            // tile_dim1 = KB  (bits 143:128)
    g1[5] = ldb;           // tensor_dim0_stride (elements)
    g1[6] = 0;
    g1[7] = 0;
    i32x4 z4 = {0, 0, 0, 0};
    i32x8 z8 = {0, 0, 0, 0, 0, 0, 0, 0};
    __builtin_amdgcn_tensor_load_to_lds(g0, g1, z4, z4, z8, 0);
#else
    (void)gsrc; (void)lds_byte; (void)ldb;
#endif
}

// ---------------------------------------------------------------------------
// Staged WMMA GEMM, C(MxN) = scale * A(MxK) @ B(KxN).
// grid = (N/16, ceil(M/64), batch); block = 128 (4 waves).
// All 4 waves (M tiles tm..tm+48) share one 16-wide B panel in LDS,
// staged KB rows at a time by the TDM (wave 0), double buffered.
// Fragment layouts per CDNA5 ISA 7.12.2.
// ---------------------------------------------------------------------------
__global__ void k_gemm_nn_s(const float* __restrict__ A, const float* __restrict__ B,
                            float* __restrict__ C, int M, int K, int lda, int ldb, int ldc,
                            int sA, int sB, int sC, float scale)
{
    __shared__ float LB[2][KB * 16];
    const int tid  = threadIdx.x;
    const int wave = tid >> 5;
    const int lane = tid & 31;
    const int half = lane >> 4;
    const int l16  = lane & 15;
    const int tn = blockIdx.x * 16;
    const int tm = blockIdx.y * 64 + wave * 16;
    A += blockIdx.z * sA;
    B += blockIdx.z * sB;
    C += blockIdx.z * sC;
    const bool active = (tm < M);
    const int row = tm + l16;
    const int nch = K / KB;

    v8f acc = {};
#if USE_TDM
    const unsigned lds0 = (unsigned)(size_t)(&LB[0][0]);
    if (wave == 0) tdm_load_tile(&B[tn], lds0, ldb);
    for (int ch = 0; ch < nch; ++ch) {
        if (wave == 0) {
            if (ch + 1 < nch) {
                tdm_load_tile(&B[(ch + 1) * KB * ldb + tn],
                              lds0 + (unsigned)(((ch + 1) & 1) * KB * 16 * 4), ldb);
                __builtin_amdgcn_s_wait_tensorcnt(1);   // oldest (chunk ch) done
            } else {
                __builtin_amdgcn_s_wait_tensorcnt(0);
            }
        }
        __syncthreads();                                 // chunk ch visible to all waves
        const float* lb = LB[ch & 1];
        if (active) {
            const int k0 = ch * KB;
#pragma unroll
            for (int kk = 0; kk < KB; kk += 4) {
                v2f a, b;
                a.x = A[row * lda + k0 + kk + 2 * half];
                a.y = A[row * lda + k0 + kk + 2 * half + 1];
                b.x = lb[(kk + 2 * half) * 16 + l16];
                b.y = lb[(kk + 2 * half + 1) * 16 + l16];
                acc = __builtin_amdgcn_wmma_f32_16x16x4_f32(false, a, false, b,
                                                            (short)0, acc, false, false);
            }
        }
        __syncthreads();                                 // done reading before overwrite
    }
#else
    for (int ch = 0; ch < nch; ++ch) {
        const int k0 = ch * KB;
        float* lb = LB[ch & 1];
        for (int e = tid; e < KB * 16; e += 128) {       // synchronous staging fallback
            int kk = e >> 4, c = e & 15;
            lb[e] = B[(k0 + kk) * ldb + tn + c];
        }
        __syncthreads();
        if (active) {
#pragma unroll
            for (int kk = 0; kk < KB; kk += 4) {
                v2f a, b;
                a.x = A[row * lda + k0 + kk + 2 * half];
                a.y = A[row * lda + k0 + kk + 2 * half + 1];
                b.x = lb[(kk + 2 * half) * 16 + l16];
                b.y = lb[(kk + 2 * half + 1) * 16 + l16];
                acc = __builtin_amdgcn_wmma_f32_16x16x4_f32(false, a, false, b,
                                                            (short)0, acc, false, false);
            }
        }
        __syncthreads();
    }
#endif
    if (active) {
#pragma unroll
        for (int v = 0; v < 8; ++v)
            C[(tm + v + 8 * half) * ldc + tn + l16] = acc[v] * scale;
    }
}

// ---------------------------------------------------------------------------
// C(MxN) = scale * A(MxK) @ B(NxK)^T  (Gram matrices, K = 16384 streaming).
// Per-lane A/B elements are K-contiguous -> b64 loads; prefetch ahead.
// grid = (N/16, M/16); block = 32 (one wave / tile).
// ---------------------------------------------------------------------------
__global__ void k_gemm_nt(const float* __restrict__ A, const float* __restrict__ B,
                          float* __restrict__ C, int K, int lda, int ldb, int ldc,
                          float scale)
{
    int lane = threadIdx.x & 31;
    int half = lane >> 4;
    int l16  = lane & 15;
    int tn = blockIdx.x * 16;
    int tm = blockIdx.y * 16;
    int row  = tm + l16;
    int colB = tn + l16;
    v8f acc = {};
    for (int k = 0; k < K; k += 4) {
        __builtin_prefetch(&A[row * lda + k + 256], 0, 1);
        __builtin_prefetch(&B[colB * ldb + k + 256], 0, 1);
        v2f a, b;
        a.x = A[row * lda + k + 2 * half];
        a.y = A[row * lda + k + 2 * half + 1];
        b.x = B[colB * ldb + k + 2 * half];      // B^T[k][n] = B[n][k]
        b.y = B[colB * ldb + k + 2 * half + 1];
        acc = __builtin_amdgcn_wmma_f32_16x16x4_f32(false, a, false, b,
                                                    (short)0, acc, false, false);
    }
#pragma unroll
    for (int v = 0; v < 8; ++v)
        C[(tm + v + 8 * half) * ldc + tn + l16] = acc[v] * scale;
}

// ---------------------------------------------------------------------------
// Adaptive median (windows 3,5,7, edge-clamped) + outlier removal + transpose
// to (band, pixel) layout.  Also writes v0 = img_median (raw, per reference).
// ---------------------------------------------------------------------------
__global__ void k_median_ro(const float* __restrict__ img, const float* __restrict__ p_thr,
                            float* __restrict__ y_og, float* __restrict__ y_med_v,
                            float* __restrict__ y_ro)
{
    int idx = blockIdx.x * blockDim.x + threadIdx.x;
    if (idx >= 128 * 128 * NB) return;
    int b = idx % NB;
    int p = idx / NB;          // p = r*128 + c
    int c0 = p & 127, r0 = p >> 7;

    float win[49];
    for (int dy = 0; dy < 7; ++dy) {
        int rr = r0 + dy - 3; rr = rr < 0 ? 0 : (rr > 127 ? 127 : rr);
        for (int dx = 0; dx < 7; ++dx) {
            int cc = c0 + dx - 3; cc = cc < 0 ? 0 : (cc > 127 ? 127 : cc);
            win[dy * 7 + dx] = img[((rr << 7) + cc) * NB + b];
        }
    }
    float x = win[3 * 7 + 3];
    float out = x, zmed_last = x;
    bool done = false;
    for (int rad = 1; rad <= 3; ++rad) {
        float buf[49];
        int m = 0;
        for (int dy = 3 - rad; dy <= 3 + rad; ++dy)
            for (int dx = 3 - rad; dx <= 3 + rad; ++dx)
                buf[m++] = win[dy * 7 + dx];
        float zmax = buf[0];
        for (int i = 1; i < m; ++i) zmax = fmaxf(zmax, buf[i]);
        int h = m >> 1;
        for (int i = 0; i <= h; ++i) {            // partial selection sort
            int mi = i;
            for (int j = i + 1; j < m; ++j) if (buf[j] < buf[mi]) mi = j;
            float t = buf[i]; buf[i] = buf[mi]; buf[mi] = t;
        }
        float zmin = buf[0], zmed = buf[h];
        bool valid = (zmin < zmed) && (zmed < zmax);
        float stage = ((zmin < x) && (x < zmax)) ? x : zmed;
        if (valid && !done) { out = stage; done = true; }
        zmed_last = zmed;
    }
    float med = done ? out : zmed_last;
    int o = b * NN + p;
    y_og[o]    = x;
    y_med_v[o] = med;                 // ADMM v0 = raw img_median
    float pt = *p_thr;
    y_ro[o] = (fabsf(x - med) > pt) ? med : x;
}

// ---------------------------------------------------------------------------
// HySime regression: Gauss-Jordan inverse of RR+eps*I, then beta rows ->
// A = I - B so that W = A @ r.
// ---------------------------------------------------------------------------
__global__ void k_aug_init(const float* __restrict__ RR, float* __restrict__ Aug)
{
    int idx = blockIdx.x * blockDim.x + threadIdx.x;
    if (idx >= NB * 384) return;
    int i = idx / 384, j = idx % 384;
    float v;
    if (j < NB)        v = RR[i * BP + j] + (i == j ? 1e-6f : 0.0f);
    else if (j < 382)  v = ((j - NB) == i) ? 1.0f : 0.0f;
    else               v = 0.0f;
    Aug[idx] = v;
}

__global__ void k_gauss_jordan(float* __restrict__ Aug)
{
    __shared__ float rowk[384];
    __shared__ float fcol[NB];
    __shared__ float piv;
    int tid = threadIdx.x;
    for (int k = 0; k < NB; ++k) {
        for (int j = tid; j < 382; j += 256) rowk[j] = Aug[k * 384 + j];
        for (int i = tid; i < NB; i += 256)  fcol[i] = Aug[i * 384 + k];
        __syncthreads();
        if (tid == 0) piv = rowk[k];
        __syncthreads();
        float ip = 1.0f / piv;
        for (int j = tid; j < 382; j += 256) rowk[j] *= ip;
        __syncthreads();
        for (int j = tid; j < 382; j += 256) Aug[k * 384 + j] = rowk[j];
        for (int i = 0; i < NB; ++i) {
            if (i == k) continue;
            float f = fcol[i];
            for (int j = tid; j < 382; j += 256)
                Aug[i * 384 + j] -= f * rowk[j];
        }
        __threadfence_block();
        __syncthreads();
    }
}

__global__ void k_extract_inv(const float* __restrict__ Aug, float* __restrict__ RRi)
{
    int idx = blockIdx.x * blockDim.x + threadIdx.x;
    if (idx >= NB * NB) return;
    int i = idx / NB, j = idx % NB;
    RRi[idx] = Aug[i * 384 + NB + j];
}

// beta[i][j] = RRi[j,:].rra - RRi[j][i]*(RRi[i,:].rra)/RRi[i][i], rra = RR[:,i] (i zeroed)
__global__ void k_beta(const float* __restrict__ RR, const float* __restrict__ RRi,
                       float* __restrict__ A_p)
{
    __shared__ float red[256];
    __shared__ float s_i;
    int i = blockIdx.x, tid = threadIdx.x;
    float part = 0.0f;
    for (int c = tid; c < NB; c += 256)
        if (c != i) part += RRi[i * NB + c] * RR[c * BP + i];
    red[tid] = part; __syncthreads();
    for (int s = 128; s > 0; s >>= 1) { if (tid < s) red[tid] += red[tid + s]; __syncthreads(); }
    if (tid == 0) s_i = red[0] / RRi[i * NB + i];
    __syncthreads();
    for (int j = tid; j < NB; j += 256) {
        float dot = 0.0f;
        for (int c = 0; c < NB; ++c)
            if (c != i) dot += RRi[j * NB + c] * RR[c * BP + i];
        float beta = dot - RRi[j * NB + i] * s_i;
        if (j == i) beta = 0.0f;
        A_p[i * BP + j] = (i == j ? 1.0f : 0.0f) - beta;
    }
}

__global__ void k_rowsumsq(const float* __restrict__ W, float* __restrict__ rw,
                           float* __restrict__ srw, float* __restrict__ hold)
{
    __shared__ float red[256];
    int i = blockIdx.x;
    float s = 0.0f;
    for (int p = threadIdx.x; p < NN; p += 256) { float w = W[i * NN + p]; s += w * w; }
    red[threadIdx.x] = s; __syncthreads();
    for (int st = 128; st > 0; st >>= 1) { if (threadIdx.x < st) red[threadIdx.x] += red[threadIdx.x + st]; __syncthreads(); }
    if (threadIdx.x == 0) {
        float v  = red[0] / (float)NN;
        float sq = sqrtf(v);
        rw[i] = v; srw[i] = sq;
        hold[i] = (sq > 1e-20f) ? 1.0f / sq : 0.0f;   // padded row -> 0
    }
}

__global__ void k_scale_rows(float* __restrict__ Y, const float* __restrict__ hold)
{
    int idx = blockIdx.x * blockDim.x + threadIdx.x;
    if (idx >= BP * NN) return;
    Y[idx] *= hold[idx >> 14];
}

// ---------------------------------------------------------------------------
// Top-10 eigenvectors of whitened covariance via orthogonal iteration + MGS.
// Sign ambiguity is irrelevant: pipeline is odd-symmetric in each eigencolumn.
// rw_fh = I  ->  sig_k = ||E[:,k]|| = 1  ->  tau = sqrt(2 ln n).
// ---------------------------------------------------------------------------
__global__ void k_subspace(const float* __restrict__ C, float* __restrict__ E_p,
                           float* __restrict__ E_T, float* __restrict__ tau, int iters)
{
    __shared__ float V[NB * 10];
    __shared__ float Wm[NB * 10];
    __shared__ float red[256];
    __shared__ float lam[10];
    __shared__ int   perm[10];
    int tid = threadIdx.x;
    for (int idx = tid; idx < NB * 10; idx += 256) {
        int i = idx / 10, c = idx % 10;
        V[idx] = sinf(0.5f + 0.37f * (float)((i + 1) * (c + 1))) + (i == c ? 2.0f : 0.0f);
    }
    __syncthreads();
    for (int it = 0; it <= iters; ++it) {
        for (int idx = tid; idx < NB * 10; idx += 256) {   // Wm = C * V
            int i = idx / 10, c = idx % 10;
            float s = 0.0f;
            for (int j = 0; j < NB; ++j) s += C[i * BP + j] * V[j * 10 + c];
            Wm[idx] = s;
        }
        __syncthreads();
        if (it == iters) break;                            // keep Wm for Rayleigh
        for (int c = 0; c < 10; ++c) {                     // modified Gram-Schmidt
            float part = 0.0f;
            for (int i = tid; i < NB; i += 256) part += Wm[i * 10 + c] * Wm[i * 10 + c];
            red[tid] = part; __syncthreads();
            for (int s = 128; s > 0; s >>= 1) { if (tid < s) red[tid] += red[tid + s]; __syncthreads(); }
            float inv = rsqrtf(red[0] + 1e-30f);
            __syncthreads();
            for (int i = tid; i < NB; i += 256) Wm[i * 10 + c] *= inv;
            __syncthreads();
            for (int c2 = c + 1; c2 < 10; ++c2) {
                part = 0.0f;
                for (int i = tid; i < NB; i += 256) part += Wm[i * 10 + c] * Wm[i * 10 + c2];
                red[tid] = part; __syncthreads();
                for (int s = 128; s > 0; s >>= 1) { if (tid < s) red[tid] += red[tid + s]; __syncthreads(); }
                float dt = red[0]; __syncthreads();
                for (int i = tid; i < NB; i += 256) Wm[i * 10 + c2] -= dt * Wm[i * 10 + c];
                __syncthreads();
            }
        }
        for (int idx = tid; idx < NB * 10; idx += 256) V[idx] = Wm[idx];
        __syncthreads();
    }
    for (int c = 0; c < 10; ++c) {                         // lam_c = V_c . (C V_c)
        float part = 0.0f;
        for (int i = tid; i < NB; i += 256) part += V[i * 10 + c] * Wm[i * 10 + c];
        red[tid] = part; __syncthreads();
        for (int s = 128; s > 0; s >>= 1) { if (tid < s) red[tid] += red[tid + s]; __syncthreads(); }
        if (tid == 0) lam[c] = red[0];
        __syncthreads();
    }
    if (tid == 0) {
        for (int c = 0; c < 10; ++c) perm[c] = c;
        for (int a = 0; a < 10; ++a) {
            int best = a;
            for (int b2 = a + 1; b2 < 10; ++b2)
                if (lam[perm[b2]] > lam[perm[best]]) best = b2;
            int t = perm[a]; perm[a] = perm[best]; perm[best] = t;
        }
    }
    __syncthreads();
    for (int idx = tid; idx < BP * KP; idx += 256) E_p[idx] = 0.0f;
    for (int idx = tid; idx < KP * BP; idx += 256) E_T[idx] = 0.0f;
    __syncthreads();
    for (int idx = tid; idx < NB * 10; idx += 256) {
        int i = idx / 10, c = idx % 10;
        float val = V[i * 10 + perm[c]];
        E_p[i * KP + c] = val;
        E_T[c * BP + i] = val;
    }
    for (int c = tid; c < KP; c += 256)
        tau[c] = (c < 10) ? sqrtf(2.0f * logf((float)NN)) : 0.0f;
}

__global__ void k_dct_init(float* __restrict__ D, float* __restrict__ Dt)
{
    int j = threadIdx.x, i = blockIdx.x;      // 128 x 128
    float c = (i == 0) ? sqrtf(1.0f / 128.0f) : sqrtf(2.0f / 128.0f);
    float v = c * cosf(PI_F * (float)(2 * j + 1) * (float)i / 256.0f);
    D[i * 128 + j]  = v;
    Dt[j * 128 + i] = v;
}

// ---------------------------------- ADMM -----------------------------------
__global__ void k_t(const float* __restrict__ yog, const float* __restrict__ v,
                    const float* __restrict__ d, float* __restrict__ S)
{
    int idx = blockIdx.x * blockDim.x + threadIdx.x;
    if (idx >= BP * NN) return;
    S[idx] = yog[idx] - v[idx] + d[idx];
}

__global__ void k_soft_rowtau(float* __restrict__ X, const float* __restrict__ tau, int n)
{
    int idx = blockIdx.x * blockDim.x + threadIdx.x;
    if (idx >= n) return;
    float thr = tau[idx >> 14];
    float x = X[idx];
    X[idx] = copysignf(fmaxf(fabsf(x) - thr, 0.0f), x);
}

__global__ void k_admm(const float* __restrict__ yog, const float* __restrict__ Ez,
                       float* __restrict__ v, float* __restrict__ d)
{
    int idx = blockIdx.x * blockDim.x + threadIdx.x;
    if (idx >= BP * NN) return;
    float yez  = yog[idx] - Ez[idx];
    float vaux = yez + d[idx];
    float vn   = copysignf(fmaxf(fabsf(vaux) - 1.0f, 0.0f), vaux);
    v[idx] = vn;
    d[idx] = vaux - vn;
}

__global__ void k_output(const float* __restrict__ S, const float* __restrict__ srw,
                         float* __restrict__ out)
{
    int idx = blockIdx.x * blockDim.x + threadIdx.x;
    if (idx >= NN * NB) return;
    int b = idx % NB, p = idx / NB;
    out[idx] = srw[b] * S[b * NN + p];
}

// ---------------------------------------------------------------------------
extern "C" void kernel_launch(void* const* d_in, const int* in_sizes, int n_in,
                              void* d_out, int out_size, void* d_ws, size_t ws_size,
                              hipStream_t stream)
{
    (void)in_sizes; (void)n_in; (void)out_size; (void)ws_size;
    const float* img   = (const float*)d_in[0];
    const float* p_thr = (const float*)d_in[2];   // d_in[1] = k_subspace (10, hardcoded)
    float* out = (float*)d_out;

    float* w = (float*)d_ws;
    size_t off = 0;
    auto alloc = [&](size_t n) { float* p = w + off; off += n; return p; };
    float* y_og = alloc((size_t)BP * NN);
    float* y_ro = alloc((size_t)BP * NN);
    float* v    = alloc((size_t)BP * NN);
    float* dbuf = alloc((size_t)BP * NN);
    float* S    = alloc((size_t)BP * NN);      // reused: W, then t / Ez
    float* z    = alloc((size_t)KP * NN);
    float* T1   = alloc((size_t)10 * NN);
    float* T2   = alloc((size_t)10 * NN);
    float* RR   = alloc(BP * BP);
    float* Aug  = alloc(NB * 384);
    float* RRi  = alloc(NB * NB);
    float* A_p  = alloc(BP * BP);
    float* Ccov = alloc(BP * BP);
    float* E_p  = alloc(BP * KP);
    float* E_T  = alloc(KP * BP);
    float* Dm   = alloc(128 * 128);
    float* Dt   = alloc(128 * 128);
    float* rw   = alloc(BP);
    float* srw  = alloc(BP);
    float* hold = alloc(BP);
    float* tau  = alloc(KP);

    // zero the padded rows / accumulator state
    (void)hipMemsetAsync(y_og + (size_t)NB * NN, 0, NN * sizeof(float), stream);
    (void)hipMemsetAsync(y_ro + (size_t)NB * NN, 0, NN * sizeof(float), stream);
    (void)hipMemsetAsync(v    + (size_t)NB * NN, 0, NN * sizeof(float), stream);
    (void)hipMemsetAsync(dbuf, 0, (size_t)BP * NN * sizeof(float), stream);
    (void)hipMemsetAsync(z,    0, (size_t)KP * NN * sizeof(float), stream);
    (void)hipMemsetAsync(A_p,  0, (size_t)BP * BP * sizeof(float), stream);

    int npix = 128 * 128 * NB;
    k_median_ro<<<(npix + 255) / 256, 256, 0, stream>>>(img, p_thr, y_og, v, y_ro);

    // --- HySime noise estimation (on raw y_ro) ---
    k_gemm_nt<<<dim3(BP / 16, BP / 16, 1), 32, 0, stream>>>(y_ro, y_ro, RR, NN, NN, NN, BP, 1.0f);
    k_aug_init<<<(NB * 384 + 255) / 256, 256, 0, stream>>>(RR, Aug);
    k_gauss_jordan<<<1, 256, 0, stream>>>(Aug);
    k_extract_inv<<<(NB * NB + 255) / 256, 256, 0, stream>>>(Aug, RRi);
    k_beta<<<NB, 256, 0, stream>>>(RR, RRi, A_p);
    k_gemm_nn_s<<<dim3(NN / 16, 3, 1), 128, 0, stream>>>(A_p, y_ro, S, BP, BP, BP, NN, NN, 0, 0, 0, 1.0f);
    k_rowsumsq<<<BP, 256, 0, stream>>>(S, rw, srw, hold);

    // --- whitening + covariance + subspace ---
    k_scale_rows<<<(BP * NN + 255) / 256, 256, 0, stream>>>(y_og, hold);
    k_scale_rows<<<(BP * NN + 255) / 256, 256, 0, stream>>>(y_ro, hold);
    k_gemm_nt<<<dim3(BP / 16, BP / 16, 1), 32, 0, stream>>>(y_ro, y_ro, Ccov, NN, NN, NN, BP, 1.0f / (float)NN);
    k_subspace<<<1, 256, 0, stream>>>(Ccov, E_p, E_T, tau, 64);
    k_dct_init<<<128, 128, 0, stream>>>(Dm, Dt);

    // --- ADMM (40 iterations) ---
    for (int it = 0; it < 40; ++it) {
        k_t<<<(BP * NN + 255) / 256, 256, 0, stream>>>(y_og, v, dbuf, S);
        // eigen_y = E^T (y - v + d)    (16 x 16384)
        k_gemm_nn_s<<<dim3(NN / 16, 1, 1), 128, 0, stream>>>(E_T, S, z, KP, BP, BP, NN, NN, 0, 0, 0, 1.0f);
        // dctn: coef = D @ X @ D^T, batched over 10 eigen-images
        k_gemm_nn_s<<<dim3(8, 2, 10), 128, 0, stream>>>(Dm, z, T1, 128, 128, 128, 128, 128, 0, NN, NN, 1.0f);
        k_gemm_nn_s<<<dim3(8, 2, 10), 128, 0, stream>>>(T1, Dt, T2, 128, 128, 128, 128, 128, NN, 0, NN, 1.0f);
        k_soft_rowtau<<<(10 * NN + 255) / 256, 256, 0, stream>>>(T2, tau, 10 * NN);
        // idctn: X = D^T @ coef @ D
        k_gemm_nn_s<<<dim3(8, 2, 10), 128, 0, stream>>>(Dt, T2, T1, 128, 128, 128, 128, 128, 0, NN, NN, 1.0f);
        k_gemm_nn_s<<<dim3(8, 2, 10), 128, 0, stream>>>(T1, Dm, z, 128, 128, 128, 128, 128, NN, 0, NN, 1.0f);
        // Ez = E @ z   (192 x 16384), then soft-threshold ADMM update
        k_gemm_nn_s<<<dim3(NN / 16, 3, 1), 128, 0, stream>>>(E_p, z, S, BP, KP, KP, NN, NN, 0, 0, 0, 1.0f);
        k_admm<<<(BP * NN + 255) / 256, 256, 0, stream>>>(y_og, S, v, dbuf);
    }

    // --- un-whiten and write (row, col, band) ---
    k_gemm_nn_s<<<dim3(NN / 16, 3, 1), 128, 0, stream>>>(E_p, z, S, BP, KP, KP, NN, NN, 0, 0, 0, 1.0f);
    k_output<<<(NN * NB + 255) / 256, 256, 0, stream>>>(S, srw, out);
}